// ContrastiveLoss_40020505264527
// MI455X (gfx1250) — compile-verified
//
#include <hip/hip_runtime.h>
#include <hip/hip_bf16.h>
#include <math.h>

// ---------------------------------------------------------------------------
// NT-Xent contrastive loss, N=4096, D=512, tau=0.1
//   sim_ij = rnorm_i * rnorm_j * (z_i . z_j) / tau   (8192x8192, never stored)
//   denom_i = sum_j exp(sim_ij) * [ (i^j)&4095 != 0 ]
//   loss = -mean_i( pos_i - log(denom_i) )
// GEMM on v_wmma_f32_16x16x4_f32; K staged via the Tensor Data Mover
// (tensor_load_to_lds, TDM LDS padding, s_wait_tensorcnt) with double
// buffering so DMA of chunk c+1 overlaps the WMMAs of chunk c.
// ---------------------------------------------------------------------------

#define NN      4096
#define DD      512
#define TWO_N   8192
#define INV_TAU 10.0f

typedef __attribute__((ext_vector_type(2))) float v2f;
typedef __attribute__((ext_vector_type(8))) float v8f;
typedef __attribute__((ext_vector_type(4))) unsigned int u32x4;
typedef __attribute__((ext_vector_type(8))) int i32x8;
typedef __attribute__((ext_vector_type(4))) int i32x4;

__device__ __forceinline__ const float* row_ptr(const float* x1, const float* x2, int r) {
    return (r < NN) ? (x1 + (size_t)r * DD) : (x2 + (size_t)(r - NN) * DD);
}

// ---- kernel 1: reciprocal row norms (wave per row) -------------------------
__global__ void cl_rownorm(const float* __restrict__ x1, const float* __restrict__ x2,
                           float* __restrict__ rnorm) {
    const int tid  = threadIdx.x;
    const int wave = tid >> 5, lane = tid & 31;
    const int r = blockIdx.x * 8 + wave;
    const float4* src = reinterpret_cast<const float4*>(row_ptr(x1, x2, r));
    float ss = 0.f;
#pragma unroll
    for (int q = 0; q < 4; ++q) {
        float4 v = src[lane + 32 * q];
        ss += v.x * v.x + v.y * v.y + v.z * v.z + v.w * v.w;
    }
    ss += __shfl_xor(ss, 1);  ss += __shfl_xor(ss, 2);
    ss += __shfl_xor(ss, 4);  ss += __shfl_xor(ss, 8);  ss += __shfl_xor(ss, 16);
    if (lane == 0) rnorm[r] = 1.0f / fmaxf(sqrtf(ss), 1e-8f);
}

// ---- kernel 2: positive-pair raw dots (wave per row, i < N) ----------------
__global__ void cl_posdot(const float* __restrict__ x1, const float* __restrict__ x2,
                          float* __restrict__ posdot) {
    const int tid  = threadIdx.x;
    const int wave = tid >> 5, lane = tid & 31;
    const int r = blockIdx.x * 8 + wave;
    const float4* a = reinterpret_cast<const float4*>(x1 + (size_t)r * DD);
    const float4* b = reinterpret_cast<const float4*>(x2 + (size_t)r * DD);
    float d = 0.f;
#pragma unroll
    for (int q = 0; q < 4; ++q) {
        float4 va = a[lane + 32 * q];
        float4 vb = b[lane + 32 * q];
        d += va.x * vb.x + va.y * vb.y + va.z * vb.z + va.w * vb.w;
    }
    d += __shfl_xor(d, 1);  d += __shfl_xor(d, 2);
    d += __shfl_xor(d, 4);  d += __shfl_xor(d, 8);  d += __shfl_xor(d, 16);
    if (lane == 0) posdot[r] = d;
}

// ---- kernel 3: TDM-fed WMMA GEMM + exp/mask/row-sum epilogue ---------------
#define KB       16        // K chunk per TDM transfer
#define LDS_STR  20        // KB + 4 pad (TDM pad_interval=16DW, pad_amount=4DW)
#define TILE_R   256
#define TILE_C   128
#define NCHUNK   (DD / KB) // 32
#define BUF_FLTS ((TILE_R + TILE_C) * LDS_STR)

// Issue one TDM 2D tile load: rows x KB fp32 elements from a row-major
// [4096 x 512] tensor into LDS with 4-DWORD padding after each 16-DWORD row.
__device__ __forceinline__ void tdm_load_tile(unsigned long long gaddr,
                                              unsigned int lds_byte, int rows) {
    u32x4 g0;
    g0[0] = 1u;                                      // count=1, user descriptor
    g0[1] = lds_byte;                                // lds_addr
    g0[2] = (unsigned int)(gaddr & 0xFFFFFFFFu);     // global_addr[31:0]
    g0[3] = (unsigned int)(gaddr >> 32) | (2u << 30);// global_addr[56:32] | type=2
    i32x8 g1;
    g1[0] = (int)((2u << 16)        // data_size = 4B
                | (1u << 20)        // pad_enable
                | (3u << 22)        // pad_interval code 3 -> 16 DWORDs
                | (3u << 25));      // pad_amount  code 3 -> 4 DWORDs
    g1[1] = (int)(512u << 16);      // tensor_dim0 = 512   (bits 79:48)
    g1[2] = (int)(4096u << 16);     // tensor_dim1 = 4096  (bits 111:80)
    g1[3] = (int)((unsigned)KB << 16); // tile_dim0 = 16   (bits 127:112)
    g1[4] = rows;                   // tile_dim1 (tile_dim2 = 0)
    g1[5] = 512;                    // tensor_dim0_stride = 512 elements
    g1[6] = 0;
    g1[7] = 0;
    i32x4 z4 = {0, 0, 0, 0};        // groups 2/3 unused (2D tile)
    i32x8 z8 = {0, 0, 0, 0, 0, 0, 0, 0};
    __builtin_amdgcn_tensor_load_to_lds(g0, g1, z4, z4, z8, 0);
}

__global__ __launch_bounds__(256)
void cl_gemm(const float* __restrict__ x1, const float* __restrict__ x2,
             const float* __restrict__ rnorm, float* __restrict__ partials) {
    __shared__ float lds[2 * BUF_FLTS];              // 61440 B, double buffered

    const int tid    = threadIdx.x;
    const int wave   = tid >> 5;
    const int lane   = tid & 31;
    const int laneLo = lane & 15;
    const int hi     = lane >> 4;
    const int rowBase = blockIdx.y * TILE_R;
    const int colBase = blockIdx.x * TILE_C;

    // LDS byte offset of the buffer (low 32 bits of generic shared address)
    const unsigned int lds_base = (unsigned int)(uintptr_t)(void*)&lds[0];
    const unsigned int bufBytes = BUF_FLTS * 4u;
    const unsigned int bOffB    = TILE_R * LDS_STR * 4u;

    // tile bases (rowBase multiple of 256, colBase of 128 -> never straddle N)
    const unsigned long long gA =
        (unsigned long long)(uintptr_t)row_ptr(x1, x2, rowBase);
    const unsigned long long gB =
        (unsigned long long)(uintptr_t)row_ptr(x1, x2, colBase);

    v8f acc[2][8];
#pragma unroll
    for (int s = 0; s < 2; ++s)
#pragma unroll
        for (int n = 0; n < 8; ++n) acc[s][n] = (v8f){};

    if (wave == 0) {                                 // prime buffer 0
        tdm_load_tile(gA, lds_base, TILE_R);
        tdm_load_tile(gB, lds_base + bOffB, TILE_C);
    }

    for (int c = 0; c < NCHUNK; ++c) {
        const float* As = &lds[(c & 1) * BUF_FLTS];
        const float* Bs = As + TILE_R * LDS_STR;

        if (wave == 0) __builtin_amdgcn_s_wait_tensorcnt(0); // chunk c landed
        __syncthreads();   // all waves done with buffer (c+1)&1, chunk c visible

        if (wave == 0 && c + 1 < NCHUNK) {           // DMA chunk c+1 overlaps
            const unsigned int pb = lds_base + (unsigned)((c + 1) & 1) * bufBytes;
            const unsigned long long off = (unsigned long long)(c + 1) * KB * 4u;
            tdm_load_tile(gA + off, pb, TILE_R);
            tdm_load_tile(gB + off, pb + bOffB, TILE_C);
        }

#pragma unroll
        for (int k0 = 0; k0 < KB; k0 += 4) {
            v2f bf[8];
#pragma unroll
            for (int n = 0; n < 8; ++n)
                bf[n] = *reinterpret_cast<const v2f*>(
                    &Bs[(16 * n + laneLo) * LDS_STR + k0 + 2 * hi]);
            v2f a0 = *reinterpret_cast<const v2f*>(
                &As[(32 * wave + laneLo) * LDS_STR + k0 + 2 * hi]);
            v2f a1 = *reinterpret_cast<const v2f*>(
                &As[(32 * wave + 16 + laneLo) * LDS_STR + k0 + 2 * hi]);
#pragma unroll
            for (int n = 0; n < 8; ++n) {
                acc[0][n] = __builtin_amdgcn_wmma_f32_16x16x4_f32(
                    false, a0, false, bf[n], (short)0, acc[0][n], false, false);
                acc[1][n] = __builtin_amdgcn_wmma_f32_16x16x4_f32(
                    false, a1, false, bf[n], (short)0, acc[1][n], false, false);
            }
        }
    }

    // epilogue: scale by rnorm_i*rnorm_j/tau, exp with self/positive mask,
    // row-sum over this 128-col tile -> unique (row, coltile) partial
    float rnCol[8];
#pragma unroll
    for (int n = 0; n < 8; ++n) rnCol[n] = rnorm[colBase + 16 * n + laneLo];

    const int rowBaseW = rowBase + 32 * wave;
#pragma unroll
    for (int s = 0; s < 2; ++s) {
#pragma unroll
        for (int v = 0; v < 8; ++v) {
            const int i = rowBaseW + 16 * s + v + 8 * hi;
            const float ri = rnorm[i] * INV_TAU;
            float partial = 0.f;
#pragma unroll
            for (int n = 0; n < 8; ++n) {
                const int j = colBase + 16 * n + laneLo;
                const float e = (((i ^ j) & (NN - 1)) == 0)
                                    ? 0.f
                                    : __expf(acc[s][n][v] * ri * rnCol[n]);
                partial += e;
            }
            partial += __shfl_xor(partial, 1);
            partial += __shfl_xor(partial, 2);
            partial += __shfl_xor(partial, 4);
            partial += __shfl_xor(partial, 8);
            if (laneLo == 0)
                partials[(size_t)i * 64 + blockIdx.x] = partial;
        }
    }
}

// ---- kernel 4: deterministic final reduction -------------------------------
__global__ void cl_finalize(const float* __restrict__ rnorm,
                            const float* __restrict__ posdot,
                            const float* __restrict__ partials,
                            float* __restrict__ out) {
    __shared__ float red[256];
    const int tid = threadIdx.x;
    float acc = 0.f;
    for (int i = tid; i < TWO_N; i += 256) {
        const float* p = partials + (size_t)i * 64;
        float denom = 0.f;
#pragma unroll
        for (int b = 0; b < 64; ++b) denom += p[b];
        const float pos = posdot[i & (NN - 1)] * rnorm[i] *
                          rnorm[(i + NN) & (TWO_N - 1)] * INV_TAU;
        acc += pos - logf(denom);
    }
    red[tid] = acc;
    __syncthreads();
#pragma unroll
    for (int s = 128; s > 0; s >>= 1) {
        if (tid < s) red[tid] += red[tid + s];
        __syncthreads();
    }
    if (tid == 0) out[0] = -red[0] / (float)TWO_N;
}

// ---------------------------------------------------------------------------
extern "C" void kernel_launch(void* const* d_in, const int* in_sizes, int n_in,
                              void* d_out, int out_size, void* d_ws, size_t ws_size,
                              hipStream_t stream) {
    (void)in_sizes; (void)n_in; (void)out_size; (void)ws_size;
    const float* x1 = (const float*)d_in[0];
    const float* x2 = (const float*)d_in[1];
    float* ws       = (float*)d_ws;

    float* rnorm    = ws;               // 8192 floats
    float* posdot   = ws + 8192;        // 4096 floats
    float* partials = ws + 12288;       // 8192 * 64 floats  (~2 MB)

    cl_rownorm<<<dim3(TWO_N / 8), dim3(256), 0, stream>>>(x1, x2, rnorm);
    cl_posdot <<<dim3(NN / 8),    dim3(256), 0, stream>>>(x1, x2, posdot);
    cl_gemm   <<<dim3(TWO_N / TILE_C, TWO_N / TILE_R), dim3(256), 0, stream>>>(
        x1, x2, rnorm, partials);
    cl_finalize<<<dim3(1), dim3(256), 0, stream>>>(rnorm, posdot, partials,
                                                   (float*)d_out);
}